// VolumeRenderer_11106785427863
// MI455X (gfx1250) — compile-verified
//
#include <hip/hip_runtime.h>
#include <stdint.h>

#define VR_NPTS 128
#define VR_FAR_DELTA 1e10f
#define WAVES_PER_BLOCK 8
#define BLOCK_THREADS 256

// CDNA5 async global->LDS path (ASYNCcnt), guarded so compile never breaks.
#if defined(__gfx1250__) && __has_builtin(__builtin_amdgcn_global_load_async_to_lds_b128) && __has_builtin(__builtin_amdgcn_s_wait_asynccnt)
#define VR_ASYNC 1
#else
#define VR_ASYNC 0
#endif

// WMMA-based wave reduction (matrix pipe is free in this memory-bound kernel).
#if defined(__gfx1250__) && __has_builtin(__builtin_amdgcn_wmma_f32_16x16x4_f32)
#define VR_WMMA 1
#else
#define VR_WMMA 0
#endif

typedef __attribute__((ext_vector_type(2))) float v2f;
typedef __attribute__((ext_vector_type(8))) float v8f;
#if VR_ASYNC
// Builtin signature (from clang diagnostic): param0 = AS1 pointer to
// int __attribute__((vector_size(16))), param1 = LDS-side pointer, then
// imm offset (applied to BOTH addresses per ISA pseudocode) and imm cpol.
typedef int vr_v4i __attribute__((vector_size(4 * sizeof(int))));
typedef __attribute__((address_space(1))) vr_v4i* vr_g_v4i;
typedef __attribute__((address_space(3))) vr_v4i* vr_l_v4i;
#endif

// Sum x over all 32 lanes, result broadcast to every lane.
// WMMA form: A = 16x4 with our values in one K-slot per lane and 0 elsewhere,
// B = all-ones 4x16  =>  D[m,n] = rowsum_m(A). Summing ALL of D (8 regs +
// xor-16 partner) gives the total sum of A's entries independent of the exact
// K/lane striping — layout-robust and exact.
__device__ __forceinline__ float vr_wave_sum(float x) {
#if VR_WMMA
  v2f a; a[0] = x;    a[1] = 0.0f;
  v2f b; b[0] = 1.0f; b[1] = 1.0f;
  v8f c = {};
  v8f d = __builtin_amdgcn_wmma_f32_16x16x4_f32(
      /*neg_a=*/false, a, /*neg_b=*/false, b,
      /*c_mod=*/(short)0, c, /*reuse_a=*/false, /*reuse_b=*/false);
  float s = d[0] + d[1] + d[2] + d[3] + d[4] + d[5] + d[6] + d[7];
  s += __shfl_xor(s, 16, 32);
  return s;
#else
  x += __shfl_xor(x, 1, 32);
  x += __shfl_xor(x, 2, 32);
  x += __shfl_xor(x, 4, 32);
  x += __shfl_xor(x, 8, 32);
  x += __shfl_xor(x, 16, 32);
  return x;
#endif
}

__global__ __launch_bounds__(BLOCK_THREADS)
void VolumeRenderer_11106785427863_kernel(
    const float* __restrict__ depth_values,  // (N, 128)
    const float* __restrict__ density,       // (N, 128)
    const float* __restrict__ feature,       // (N, 128, 3)
    float* __restrict__ out_feat,            // (N, 3)
    float* __restrict__ out_depth,           // (N,)
    int n_rays)
{
#if VR_ASYNC
  __shared__ float feat_lds[WAVES_PER_BLOCK][VR_NPTS * 3];  // 12 KB / block
#endif
  const int tid  = (int)threadIdx.x;
  const int wv   = tid >> 5;
  const int lane = tid & 31;
  const int ray  = (int)blockIdx.x * WAVES_PER_BLOCK + wv;
  if (ray >= n_rays) return;   // whole-wave predicate: EXEC stays all-ones

  const size_t rbase = (size_t)ray * VR_NPTS;
  // Each lane owns samples [4*lane, 4*lane+3]: coalesced b128 streams.
  const float4 dep = ((const float4*)(depth_values + rbase))[lane];
  const float4 den = ((const float4*)(density + rbase))[lane];
  const float* fsrc = feature + rbase * 3 + (size_t)lane * 12;  // 48B/lane

#if VR_ASYNC
  {
    // Kick the bulky feature stream (60% of bytes) down the async path so it
    // overlaps with the transmittance scan below (ASYNCcnt vs LOADcnt).
    float* ldst = &feat_lds[wv][lane * 12];
    vr_g_v4i g = (vr_g_v4i)fsrc;
    vr_l_v4i l = (vr_l_v4i)ldst;
    __builtin_amdgcn_global_load_async_to_lds_b128(g, l, 0, 0);
    __builtin_amdgcn_global_load_async_to_lds_b128(g, l, 16, 0);
    __builtin_amdgcn_global_load_async_to_lds_b128(g, l, 32, 0);
  }
#endif

  // deltas: need first depth of next lane for the 4th segment.
  const float nxt = __shfl_down(dep.x, 1, 32);
  const float t0 = (dep.y - dep.x) * den.x;
  const float t1 = (dep.z - dep.y) * den.y;
  const float t2 = (dep.w - dep.z) * den.z;
  const float d3 = (lane == 31) ? VR_FAR_DELTA : (nxt - dep.w);
  const float t3 = d3 * den.w;

  // lane-local inclusive prefix of tau
  const float s0 = t0, s1 = s0 + t1, s2 = s1 + t2, s3 = s2 + t3;

  // wave32 inclusive scan of lane totals
  float run = s3;
  {
    float v;
    v = __shfl_up(run, 1, 32);  if (lane >= 1)  run += v;
    v = __shfl_up(run, 2, 32);  if (lane >= 2)  run += v;
    v = __shfl_up(run, 4, 32);  if (lane >= 4)  run += v;
    v = __shfl_up(run, 8, 32);  if (lane >= 8)  run += v;
    v = __shfl_up(run, 16, 32); if (lane >= 16) run += v;
  }
  const float excl = run - s3;  // exclusive prefix entering this lane

  // weights via telescoping transmittance: w_j = T_j - T_{j+1} = T_j * alpha_j
  const float T0 = __expf(-excl);
  const float E0 = __expf(-(excl + s0));
  const float E1 = __expf(-(excl + s1));
  const float E2 = __expf(-(excl + s2));
  const float E3 = __expf(-(excl + s3));
  const float w0 = T0 - E0, w1 = E0 - E1, w2 = E1 - E2, w3 = E2 - E3;

  // features: 12 floats per lane (4 samples x RGB)
  float4 q0, q1, q2;
#if VR_ASYNC
  __builtin_amdgcn_s_wait_asynccnt(0);
  const float4* lq = (const float4*)&feat_lds[wv][lane * 12];
  q0 = lq[0]; q1 = lq[1]; q2 = lq[2];
#else
  const float4* gq = (const float4*)fsrc;
  q0 = gq[0]; q1 = gq[1]; q2 = gq[2];
#endif

  // sample j color layout: s0=(q0.x,q0.y,q0.z) s1=(q0.w,q1.x,q1.y)
  //                        s2=(q1.z,q1.w,q2.x) s3=(q2.y,q2.z,q2.w)
  float fx = w0 * q0.x + w1 * q0.w + w2 * q1.z + w3 * q2.y;
  float fy = w0 * q0.y + w1 * q1.x + w2 * q1.w + w3 * q2.z;
  float fz = w0 * q0.z + w1 * q1.y + w2 * q2.x + w3 * q2.w;
  float dd = w0 * dep.x + w1 * dep.y + w2 * dep.z + w3 * dep.w;

  fx = vr_wave_sum(fx);
  fy = vr_wave_sum(fy);
  fz = vr_wave_sum(fz);
  dd = vr_wave_sum(dd);

  if (lane == 0) {
    out_feat[3 * (size_t)ray + 0] = fx;
    out_feat[3 * (size_t)ray + 1] = fy;
    out_feat[3 * (size_t)ray + 2] = fz;
    out_depth[ray] = dd;
  }
}

extern "C" void kernel_launch(void* const* d_in, const int* in_sizes, int n_in,
                              void* d_out, int out_size, void* d_ws, size_t ws_size,
                              hipStream_t stream) {
  (void)n_in; (void)out_size; (void)d_ws; (void)ws_size;
  const float* depth   = (const float*)d_in[0];
  const float* density = (const float*)d_in[1];
  const float* feature = (const float*)d_in[2];
  const int n_rays = in_sizes[0] / VR_NPTS;

  float* out_feat  = (float*)d_out;                      // (N,3) first
  float* out_depth = out_feat + (size_t)3 * n_rays;      // then (N,)

  const int blocks = (n_rays + WAVES_PER_BLOCK - 1) / WAVES_PER_BLOCK;
  VolumeRenderer_11106785427863_kernel<<<dim3(blocks), dim3(BLOCK_THREADS), 0, stream>>>(
      depth, density, feature, out_feat, out_depth, n_rays);
}